// WeightedRuleLayer_33517924778459
// MI455X (gfx1250) — compile-verified
//
#include <hip/hip_runtime.h>
#include <hip/hip_bf16.h>

typedef __attribute__((ext_vector_type(2))) float v2f;
typedef __attribute__((ext_vector_type(8))) float v8f;

#if __has_builtin(__builtin_amdgcn_tanhf)
__device__ __forceinline__ float fast_tanh(float x) { return __builtin_amdgcn_tanhf(x); }
#else
__device__ __forceinline__ float fast_tanh(float x) { return tanhf(x); }
#endif

// y[b,d] = tanh( sum_w values[idx[w,b]][d] * weights[w][d] ),  D=16, W=8.
// One wave32 per 16-row tile of b. Exact-fp32 WMMA:
//   C[16b x 16d] += A[16 x 4] * Bm[4 x 16] chained over 32 K-chunks,
// where chunk (w, d0) covers contraction slots k=0..3 <-> (w, d0+k):
//   A[m,k]  = values[idx[w, btile+m]][d0+k]          (the gather itself)
//   Bm[k,n] = (n == d0+k) ? weights[w][n] : 0        (block-diagonal)
__global__ __launch_bounds__(256) void rule_layer_wmma_f32(
    const float* __restrict__ values,   // [N_SRC, 16]
    const float* __restrict__ weights,  // [8, 16]
    const int*   __restrict__ indices,  // [8, B]
    float*       __restrict__ out,      // [B, 16]
    int Bn, int ntiles)
{
  const int lane = threadIdx.x & 31;
  const int n    = lane & 15;   // output column d / weight column
  const int hi   = lane >> 4;   // half-wave: 0 -> K=0,1 & M=0..7 ; 1 -> K=2,3 & M=8..15
  const int koff = hi * 2;

  const int waveId = blockIdx.x * (blockDim.x >> 5) + (threadIdx.x >> 5);
  if (waveId >= ntiles) return;            // uniform per wave: EXEC stays all-ones
  const int btile = waveId * 16;

  // Per-lane weight column: wl[w] = weights[w][n]
  float wl[8];
#pragma unroll
  for (int w = 0; w < 8; ++w) wl[w] = weights[w * 16 + n];

  // Row indices for this tile (mirrored across both half-waves; same cachelines)
  int b = btile + n;
  if (b >= Bn) b = Bn - 1;                  // clamp: tail rows gather row safely
  int rows[8];
#pragma unroll
  for (int w = 0; w < 8; ++w) rows[w] = indices[(size_t)w * (size_t)Bn + b];

  v8f acc = {};
#pragma unroll
  for (int w = 0; w < 8; ++w) {
    // lane reads values[row] + koff (+d0): float2 per chunk -> exact A layout
    const v2f* pw = (const v2f*)(values + (size_t)rows[w] * 16 + koff);
#pragma unroll
    for (int c = 0; c < 4; ++c) {
      const int d0 = c * 4;
      v2f a = pw[c * 2];                    // { [d0+koff], [d0+koff+1] }
      v2f bm;
      bm.x = (n == d0 + koff)     ? wl[w] : 0.0f;   // B row K=koff
      bm.y = (n == d0 + koff + 1) ? wl[w] : 0.0f;   // B row K=koff+1
      acc = __builtin_amdgcn_wmma_f32_16x16x4_f32(
          false, a, false, bm, (short)0, acc, false, false);
    }
  }

  // C/D layout: VGPR r holds row M = r + 8*hi, column N = n.
  float* op = out + (size_t)(btile + hi * 8) * 16 + n;
#pragma unroll
  for (int r = 0; r < 8; ++r) {
    int rowb = btile + hi * 8 + r;
    float v = fast_tanh(acc[r]);
    if (rowb < Bn) op[(size_t)r * 16] = v;  // divergence only after all WMMAs
  }
}

extern "C" void kernel_launch(void* const* d_in, const int* in_sizes, int n_in,
                              void* d_out, int out_size, void* d_ws, size_t ws_size,
                              hipStream_t stream) {
  const float* values  = (const float*)d_in[0];
  const float* weights = (const float*)d_in[1];
  const int*   indices = (const int*)d_in[2];
  float*       out     = (float*)d_out;

  const int W  = 8;
  const int Bn = in_sizes[2] / W;           // 500000
  const int ntiles = (Bn + 15) / 16;        // 31250
  const int wavesPerBlock = 256 / 32;
  const int blocks = (ntiles + wavesPerBlock - 1) / wavesPerBlock;

  rule_layer_wmma_f32<<<blocks, 256, 0, stream>>>(values, weights, indices, out,
                                                  Bn, ntiles);
}